// ResidualVectorQuantizer_24206435680855
// MI455X (gfx1250) — compile-verified
//
#include <hip/hip_runtime.h>

typedef float v2f __attribute__((ext_vector_type(2)));
typedef float v8f __attribute__((ext_vector_type(8)));

#define TOKENS   32768          // 8 * 4096
#define IN_CH    512
#define EMB_CH   256
#define NCODES   1024
#define VQN      3
#define ZQ_ELEMS (TOKENS * IN_CH)   // 16,777,216

#define KCH   64                // K-chunk staged in LDS per pass
#define BROWS 128               // 8 n-tiles * 16 rows staged per block
#define BPAD  68                // LDS row stride in floats: bank = 4r+koff, conflict-free

// D = A(16x4 f32) * B(4x16 f32) + C(16x16 f32)  -- CDNA5 V_WMMA_F32_16X16X4_F32
__device__ __forceinline__ v8f wmma4(v2f a, v2f b, v8f c) {
    return __builtin_amdgcn_wmma_f32_16x16x4_f32(false, a, false, b, (short)0, c,
                                                 false, false);
}

// Cooperative async stage of a 128-row x 64-float weight chunk into LDS.
// 2048 16-byte pieces, 8 per thread. ASYNCcnt-tracked; caller barriers after.
__device__ __forceinline__ void stage_b_async(float* sB, const float* __restrict__ gsrc,
                                              int row_stride, int tid) {
#pragma unroll
    for (int j = 0; j < 8; ++j) {
        int c   = tid + 256 * j;
        int row = c >> 4;
        int col = (c & 15) << 2;                       // float offset within 64
        unsigned lofs  = (unsigned)(unsigned long long)(void*)&sB[row * BPAD + col];
        const float* g = gsrc + (size_t)row * row_stride + col;
        asm volatile("global_load_async_to_lds_b128 %0, %1, off"
                     :: "v"(lofs), "v"(g) : "memory");
    }
    asm volatile("s_wait_asynccnt 0x0" ::: "memory");  // this wave's copies done
}

// ---------------------------------------------------------------- init
__global__ void k_zero_zq(float* zq) {
    for (int i = blockIdx.x * blockDim.x + threadIdx.x; i < ZQ_ELEMS;
         i += gridDim.x * blockDim.x)
        zq[i] = 0.0f;
}

__global__ void k_vnorm(const float* __restrict__ cb, float* vnorm, float* acc) {
    int t = blockIdx.x * blockDim.x + threadIdx.x;
    if (t < 4) acc[t] = 0.0f;                 // L0,L1,L2, sumsq
    if (t < VQN * NCODES) {
        const float* v = cb + t * EMB_CH;
        float s = 0.0f;
        for (int k = 0; k < EMB_CH; ++k) { float x = v[k]; s += x * x; }
        vnorm[t] = s;
    }
}

// ------------------------------------------------ zi = (z - z_q) @ W_in^T
// Block: 8 waves = 8 m-tiles, all sharing one 128-row n-group of W_in staged
// in LDS. Each wave: 8 N-tiles in registers (64 acc VGPRs), 8 WMMA per A-load.
__global__ void k_gemm_in(const float* __restrict__ z, const float* __restrict__ zq,
                          const float* __restrict__ Win, float* __restrict__ zi) {
    __shared__ float sB[BROWS * BPAD];        // 34 KB
    int tid  = threadIdx.x;
    int wave = tid >> 5, lane = tid & 31;
    int mg = blockIdx.x >> 1;                 // 0..255
    int ng = blockIdx.x & 1;                  // 0..1  (N=256 -> 2 groups of 128)
    int mtile = mg * 8 + wave;                // 0..2047
    int rc = lane & 15, koff = (lane >> 4) << 1, hi = lane >> 4;
    int nbase = ng * BROWS;

    const float* za  = z  + (size_t)(mtile * 16 + rc) * IN_CH + koff;
    const float* zqa = zq + (size_t)(mtile * 16 + rc) * IN_CH + koff;

    v8f acc[8] = {};
    for (int kc = 0; kc < IN_CH; kc += KCH) {
        stage_b_async(sB, Win + (size_t)nbase * IN_CH + kc, IN_CH, tid);
        __syncthreads();
        const float* bp = &sB[rc * BPAD + koff];
        for (int kk = 0; kk < KCH; kk += 4) {
            v2f a;
            a.x = za[kc + kk]     - zqa[kc + kk];
            a.y = za[kc + kk + 1] - zqa[kc + kk + 1];
#pragma unroll
            for (int t = 0; t < 8; ++t) {
                v2f b;
                b.x = bp[t * 16 * BPAD + kk];
                b.y = bp[t * 16 * BPAD + kk + 1];
                acc[t] = wmma4(a, b, acc[t]);
            }
        }
        __syncthreads();
    }
#pragma unroll
    for (int t = 0; t < 8; ++t) {
        float* outp = zi + (size_t)(mtile * 16) * EMB_CH + nbase + t * 16 + rc;
#pragma unroll
        for (int j = 0; j < 8; ++j) outp[(j + 8 * hi) * EMB_CH] = acc[t][j];
    }
}

// --------- distances + argmin over 1024 codes; one wave per 16-token tile.
// dist = ||v||^2 - 2*zi.v  (||zi||^2 is row-constant for argmin; added back
// only for the commitment-loss sum). 8 score tiles per pass, A reused 8x.
__global__ void k_argmin(const float* __restrict__ zi, const float* __restrict__ cb,
                         const float* __restrict__ vnorm, int* __restrict__ idxb,
                         float* lossAcc) {
    __shared__ float smin[8][16];
    __shared__ int   sidx[8][16];
    int wave  = threadIdx.x >> 5;             // 0..7
    int lane  = threadIdx.x & 31;
    int mtile = blockIdx.x * 8 + wave;        // 0..2047
    int rc    = lane & 15;
    int koff  = (lane >> 4) << 1;
    const float* A = zi + (size_t)(mtile * 16 + rc) * EMB_CH + koff;

    float minv[8];
    int   mini[8];
#pragma unroll
    for (int j = 0; j < 8; ++j) { minv[j] = 3.4e38f; mini[j] = 0; }
    float na = 0.0f;                           // partial ||zi_row||^2

    for (int sg = 0; sg < 8; ++sg) {           // 8 code-tiles per pass
        v8f acc[8] = {};
        const float* B0 = cb + (size_t)((sg * 8) * 16 + rc) * EMB_CH + koff;
        for (int k = 0; k < EMB_CH; k += 4) {
            v2f a; a.x = A[k]; a.y = A[k + 1];
            if (sg == 0) na += a.x * a.x + a.y * a.y;
#pragma unroll
            for (int t = 0; t < 8; ++t) {
                v2f b;
                b.x = B0[t * 16 * EMB_CH + k];
                b.y = B0[t * 16 * EMB_CH + k + 1];
                acc[t] = wmma4(a, b, acc[t]);
            }
        }
#pragma unroll
        for (int t = 0; t < 8; ++t) {
            int   s  = (sg * 8 + t) * 16 + rc;
            float vn = vnorm[s];
#pragma unroll
            for (int j = 0; j < 8; ++j) {
                float d = vn - 2.0f * acc[t][j];
                if (d < minv[j]) { minv[j] = d; mini[j] = s; }  // first-min ties
            }
        }
    }

    // reduce (min, argmin) across the 16-lane half-wave holding each row
#pragma unroll
    for (int j = 0; j < 8; ++j) {
#pragma unroll
        for (int m = 8; m >= 1; m >>= 1) {
            float ov = __shfl_xor(minv[j], m, 32);
            int   oi = __shfl_xor(mini[j], m, 32);
            if (ov < minv[j] || (ov == minv[j] && oi < mini[j])) {
                minv[j] = ov; mini[j] = oi;
            }
        }
    }
    na += __shfl_xor(na, 16, 32);              // full ||zi_row||^2 on both halves

    if (rc == 0) {                             // lanes 0 and 16 own rows 0-7 / 8-15
#pragma unroll
        for (int j = 0; j < 8; ++j) {
            smin[wave][(lane >> 4) * 8 + j] = minv[j];
            sidx[wave][(lane >> 4) * 8 + j] = mini[j];
        }
    }
    // same-wave LDS RAW; DS ops are in-order per wave
    float contrib = 0.0f;
    if (lane < 16) {
        int m = mtile * 16 + lane;
        idxb[m] = sidx[wave][lane];
        contrib = smin[wave][lane] + na;       // ||zi - zq||^2 for this token
    }
#pragma unroll
    for (int m2 = 16; m2 >= 1; m2 >>= 1) contrib += __shfl_xor(contrib, m2, 32);
    if (lane == 0) atomicAdd(lossAcc, contrib);
}

// ------------------- z_q += cb[idx] @ W_out^T  (gather fused into A-load)
// Same blocking as k_gemm_in: 8 waves x 8 N-tiles, W_out chunk in LDS.
__global__ void k_gemm_out(const float* __restrict__ cb, const int* __restrict__ idxb,
                           const float* __restrict__ Wout, float* __restrict__ zq) {
    __shared__ float sB[BROWS * BPAD];
    int tid  = threadIdx.x;
    int wave = tid >> 5, lane = tid & 31;
    int mg = blockIdx.x >> 2;                 // 0..255
    int ng = blockIdx.x & 3;                  // 0..3  (N=512 -> 4 groups of 128)
    int mtile = mg * 8 + wave;
    int rc = lane & 15, koff = (lane >> 4) << 1, hi = lane >> 4;
    int nbase = ng * BROWS;

    int s = idxb[mtile * 16 + rc];            // codebook row for this token's row
    const float* A = cb + (size_t)s * EMB_CH + koff;

    v8f acc[8] = {};
    for (int kc = 0; kc < EMB_CH; kc += KCH) {
        stage_b_async(sB, Wout + (size_t)nbase * EMB_CH + kc, EMB_CH, tid);
        __syncthreads();
        const float* bp = &sB[rc * BPAD + koff];
        for (int kk = 0; kk < KCH; kk += 4) {
            v2f a;
            a.x = A[kc + kk];
            a.y = A[kc + kk + 1];
#pragma unroll
            for (int t = 0; t < 8; ++t) {
                v2f b;
                b.x = bp[t * 16 * BPAD + kk];
                b.y = bp[t * 16 * BPAD + kk + 1];
                acc[t] = wmma4(a, b, acc[t]);
            }
        }
        __syncthreads();
    }
#pragma unroll
    for (int t = 0; t < 8; ++t) {
        float* outp = zq + (size_t)(mtile * 16) * IN_CH + nbase + t * 16 + rc;
#pragma unroll
        for (int j = 0; j < 8; ++j) outp[(j + 8 * hi) * IN_CH] += acc[t][j];
    }
}

// ---------------------------------------------------- final reductions
__global__ void k_sumsq(const float* __restrict__ z, const float* __restrict__ zq,
                        float* acc) {
    __shared__ float red[256];
    float s = 0.0f;
    for (int i = blockIdx.x * blockDim.x + threadIdx.x; i < ZQ_ELEMS;
         i += gridDim.x * blockDim.x) {
        float d = zq[i] - z[i];
        s += d * d;
    }
    red[threadIdx.x] = s;
    __syncthreads();
    for (int off = 128; off > 0; off >>= 1) {
        if (threadIdx.x < off) red[threadIdx.x] += red[threadIdx.x + off];
        __syncthreads();
    }
    if (threadIdx.x == 0) atomicAdd(&acc[3], red[0]);
}

__global__ void k_finalize(const int* __restrict__ idxb, const float* __restrict__ acc,
                           float* __restrict__ out) {
    int t = blockIdx.x * blockDim.x + threadIdx.x;
    if (t < TOKENS) {
        long v = (long)idxb[t] + 1024l * idxb[TOKENS + t]
               + 1048576l * idxb[2 * TOKENS + t];
        out[ZQ_ELEMS + t] = (float)v;          // index_sum, flat after z_q
    }
    if (t == 0) {
        // per-stage loss = 2*sum||zq-zi||^2 / (TOKENS*EMB_CH); avg over 3 stages
        float stage = (acc[0] + acc[1] + acc[2]) * 2.0f / (float)(TOKENS * EMB_CH);
        float loss  = stage / 3.0f + acc[3] / (float)ZQ_ELEMS;
        out[ZQ_ELEMS + TOKENS] = loss;
    }
}

extern "C" void kernel_launch(void* const* d_in, const int* in_sizes, int n_in,
                              void* d_out, int out_size, void* d_ws, size_t ws_size,
                              hipStream_t stream) {
    (void)in_sizes; (void)n_in; (void)out_size; (void)ws_size;
    const float* z    = (const float*)d_in[0];   // [8,4096,512]
    const float* Win  = (const float*)d_in[1];   // [3,256,512]
    const float* Wout = (const float*)d_in[2];   // [3,512,256]
    const float* cb   = (const float*)d_in[3];   // [3,1024,256]

    float* out = (float*)d_out;
    float* zq  = out;                            // z_q lives in d_out[0 .. ZQ_ELEMS)

    float* ws    = (float*)d_ws;
    float* zi    = ws;                           // 8,388,608 f32 (32 MB)
    float* vnorm = ws + (size_t)TOKENS * EMB_CH; // 3*1024 f32
    int*   idxb  = (int*)(vnorm + VQN * NCODES); // 3*32768 i32
    float* acc   = (float*)(idxb + VQN * TOKENS);// 4 f32

    k_zero_zq<<<2048, 256, 0, stream>>>(zq);
    k_vnorm<<<12, 256, 0, stream>>>(cb, vnorm, acc);

    for (int i = 0; i < VQN; ++i) {
        const float* Wi  = Win  + (size_t)i * EMB_CH * IN_CH;
        const float* Wo  = Wout + (size_t)i * IN_CH * EMB_CH;
        const float* cbi = cb   + (size_t)i * NCODES * EMB_CH;
        k_gemm_in<<<512, 256, 0, stream>>>(z, zq, Wi, zi);                    // 4096 waves
        k_argmin<<<256, 256, 0, stream>>>(zi, cbi, vnorm + i * NCODES,
                                          idxb + i * TOKENS, acc + i);        // 2048 waves
        k_gemm_out<<<1024, 256, 0, stream>>>(cbi, idxb + i * TOKENS, Wo, zq); // 8192 waves
    }

    k_sumsq<<<2048, 256, 0, stream>>>(z, zq, acc);
    k_finalize<<<128, 256, 0, stream>>>(idxb, acc, out);
}